// DrugGAT_71244917506338
// MI455X (gfx1250) — compile-verified
//
#include <hip/hip_runtime.h>
#include <hip/hip_bf16.h>
#include <math.h>

// Problem constants (from reference)
#define NN   10000
#define EE   320000
#define ETOT (EE + NN)   // edges + self loops
#define HC   256

typedef __attribute__((ext_vector_type(2))) float v2f;
typedef __attribute__((ext_vector_type(8))) float v8f;

// ---------------------------------------------------------------- utilities

__global__ __launch_bounds__(256) void k_fill(float* __restrict__ p, int n, float v) {
    int t = blockIdx.x * blockDim.x + threadIdx.x;
    if (t < n) p[t] = v;
}

__global__ __launch_bounds__(256) void k_fill_int(int* __restrict__ p, int n, int v) {
    int t = blockIdx.x * blockDim.x + threadIdx.x;
    if (t < n) p[t] = v;
}

__global__ __launch_bounds__(256) void k_prep_edges(const long long* __restrict__ ei,
                                                    int* __restrict__ s,
                                                    int* __restrict__ d) {
    int i = blockIdx.x * blockDim.x + threadIdx.x;
    if (i >= ETOT) return;
    if (i < EE) { s[i] = (int)ei[i]; d[i] = (int)ei[EE + i]; }
    else        { s[i] = i - EE;     d[i] = i - EE; }
}

// ------------------------------------------------- CSR build (group edges by dst)

__global__ __launch_bounds__(256) void k_deg(const int* __restrict__ dst32,
                                             int* __restrict__ deg) {
    int i = blockIdx.x * blockDim.x + threadIdx.x;
    if (i >= ETOT) return;
    atomicAdd(&deg[dst32[i]], 1);
}

// single block, 256 threads: exclusive scan of deg[0..NN-1] -> rowptr, cursor
__global__ __launch_bounds__(256) void k_scan(const int* __restrict__ deg,
                                              int* __restrict__ rowptr,
                                              int* __restrict__ cursor) {
    __shared__ int part[256];
    const int t = threadIdx.x;
    const int CHUNK = 40;                       // 256*40 = 10240 >= NN
    int base = t * CHUNK;
    int local[CHUNK];
    int s = 0;
    for (int i = 0; i < CHUNK; ++i) {
        int idx = base + i;
        int v = (idx < NN) ? deg[idx] : 0;
        local[i] = s;                            // exclusive within chunk
        s += v;
    }
    part[t] = s;
    __syncthreads();
    // Hillis-Steele inclusive scan over 256 partials
    for (int off = 1; off < 256; off <<= 1) {
        int v = (t >= off) ? part[t - off] : 0;
        __syncthreads();
        part[t] += v;
        __syncthreads();
    }
    int offset = (t > 0) ? part[t - 1] : 0;
    for (int i = 0; i < CHUNK; ++i) {
        int idx = base + i;
        if (idx < NN) {
            int r = offset + local[i];
            rowptr[idx] = r;
            cursor[idx] = r;
        }
    }
    if (t == 255) rowptr[NN] = offset + s;       // == ETOT
}

__global__ __launch_bounds__(256) void k_scatter(const int* __restrict__ src32,
                                                 const int* __restrict__ dst32,
                                                 int* __restrict__ cursor,
                                                 int* __restrict__ esrc) {
    int i = blockIdx.x * blockDim.x + threadIdx.x;
    if (i >= ETOT) return;
    int pos = atomicAdd(&cursor[dst32[i]], 1);
    esrc[pos] = src32[i];
}

// Pack W[K,Mout] (row-major) into V_WMMA_F32_16X16X4_F32 B-operand lane order:
// tile (kt,nt) -> 32 lanes x 2 floats contiguous; lane holds
// B[kt*4 + 2*(lane>>4) + j][nt*16 + (lane&15)], j in {0,1}.
__global__ __launch_bounds__(256) void k_pack_w(const float* __restrict__ W,
                                                float* __restrict__ Wp,
                                                int K, int Mout) {
    int t = blockIdx.x * blockDim.x + threadIdx.x;
    if (t >= K * Mout) return;
    int tile = t >> 6, rem = t & 63, lane = rem >> 1, j = rem & 1;
    int colTiles = Mout >> 4;
    int kt = tile / colTiles, nt = tile - kt * colTiles;
    int k = (kt << 2) + ((lane >> 4) << 1) + j;
    int n = (nt << 4) + (lane & 15);
    Wp[t] = W[(size_t)k * Mout + n];
}

// ------------------------------------------------- WMMA f32 GEMM: Y = act(X@W + b)
// X: [nrows, ldx] row-major f32, Wp: packed B tiles, Y: [nrows, ldy].
// Each wave computes a 16x16 tile of Y. act: 0=none, 1=relu.
__global__ __launch_bounds__(256) void k_gemm_wmma(const float* __restrict__ X, int ldx,
                                                   const float* __restrict__ Wp,
                                                   const float* __restrict__ bias,
                                                   float* __restrict__ Y, int ldy,
                                                   int nrows, int K, int Mout, int act) {
    const int lane = threadIdx.x & 31;
    const int wave = blockIdx.x * (blockDim.x >> 5) + (threadIdx.x >> 5);
    const int colTiles = Mout >> 4;
    const int rowTiles = (nrows + 15) >> 4;
    if (wave >= rowTiles * colTiles) return;           // wave-uniform: EXEC stays full
    const int rowTile = wave / colTiles;
    const int colTile = wave - rowTile * colTiles;
    const int rb   = rowTile << 4;
    const int half = lane >> 4;
    const int nn   = lane & 15;

    int arow = rb + nn; if (arow > nrows - 1) arow = nrows - 1;   // clamp, keep EXEC full
    const float* xrow  = X  + (size_t)arow * ldx + (half << 1);
    const float* wptr  = Wp + ((size_t)colTile << 6) + (lane << 1);
    const size_t wstep = (size_t)colTiles << 6;

    float bv = bias ? bias[(colTile << 4) + nn] : 0.0f;
    v8f c;
#pragma unroll
    for (int r = 0; r < 8; ++r) c[r] = bv;

    const int ktiles = K >> 2;
    for (int kt = 0; kt < ktiles; ++kt) {
        v2f a = *(const v2f*)(xrow + (kt << 2));
        v2f b = *(const v2f*)(wptr + (size_t)kt * wstep);
        c = __builtin_amdgcn_wmma_f32_16x16x4_f32(false, a, false, b, (short)0, c,
                                                  false, false);
    }

    const int col = (colTile << 4) + nn;
#pragma unroll
    for (int r = 0; r < 8; ++r) {
        int row = rb + (half << 3) + r;
        if (row < nrows) {
            float v = c[r];
            if (act == 1) v = v > 0.0f ? v : 0.0f;
            Y[(size_t)row * ldy + col] = v;
        }
    }
}

// ------------------------------------------------- attention scalars per (node, head)
__global__ __launch_bounds__(256) void k_alpha(const float* __restrict__ xl,
                                               const float* __restrict__ att_src,
                                               const float* __restrict__ att_dst,
                                               float* __restrict__ as_,
                                               float* __restrict__ ad_) {
    int t = blockIdx.x * blockDim.x + threadIdx.x;   // n*4 + h
    if (t >= NN * 4) return;
    int n = t >> 2, hh = t & 3;
    const float* v  = xl + (size_t)n * HC + (hh << 6);
    const float* ws = att_src + (hh << 6);
    const float* wd = att_dst + (hh << 6);
    float s = 0.0f, d = 0.0f;
#pragma unroll 4
    for (int c = 0; c < 64; ++c) { s += v[c] * ws[c]; d += v[c] * wd[c]; }
    as_[t] = s; ad_[t] = d;
}

// ------------------------------------------------- fused GAT aggregation + epilogue
// One wave32 per destination node:
//  - segment softmax (max/sum) over in-edges via wave reductions, no atomics
//  - weighted feature gather: lanes own the 256 channels (8 regs x 32 lanes),
//    serial loop over in-edges with coalesced 1KB gathers from xl[src]
//  - fused bias + ELU + residual + LayerNorm, in-place update of h
__global__ __launch_bounds__(256) void k_gat_node(const float* __restrict__ xl,
                                                  const float* __restrict__ as_,
                                                  const float* __restrict__ ad_,
                                                  const int* __restrict__ rowptr,
                                                  const int* __restrict__ esrc,
                                                  const float* __restrict__ gat_b,
                                                  const float* __restrict__ ln_g,
                                                  const float* __restrict__ ln_b,
                                                  float* __restrict__ h) {
    const int n    = blockIdx.x * (blockDim.x >> 5) + (threadIdx.x >> 5);
    const int lane = threadIdx.x & 31;
    if (n >= NN) return;
    const int beg = rowptr[n];
    const int end = rowptr[n + 1];

    float adv[4];
#pragma unroll
    for (int hh = 0; hh < 4; ++hh) adv[hh] = ad_[(n << 2) + hh];

    // phase 1: per-head max over in-edges
    float mx[4] = {-INFINITY, -INFINITY, -INFINITY, -INFINITY};
    for (int j = beg + lane; j < end; j += 32) {
        int s = esrc[j];
#pragma unroll
        for (int hh = 0; hh < 4; ++hh) {
            float e = as_[(s << 2) + hh] + adv[hh];
            e = e > 0.0f ? e : 0.2f * e;                 // leaky_relu(0.2)
            mx[hh] = fmaxf(mx[hh], e);
        }
    }
#pragma unroll
    for (int hh = 0; hh < 4; ++hh)
        for (int off = 16; off > 0; off >>= 1)
            mx[hh] = fmaxf(mx[hh], __shfl_xor(mx[hh], off, 32));

    // phase 2: per-head exp-sum
    float sm[4] = {0.0f, 0.0f, 0.0f, 0.0f};
    for (int j = beg + lane; j < end; j += 32) {
        int s = esrc[j];
#pragma unroll
        for (int hh = 0; hh < 4; ++hh) {
            float e = as_[(s << 2) + hh] + adv[hh];
            e = e > 0.0f ? e : 0.2f * e;
            sm[hh] += __expf(e - mx[hh]);
        }
    }
#pragma unroll
    for (int hh = 0; hh < 4; ++hh) {
        for (int off = 16; off > 0; off >>= 1)
            sm[hh] += __shfl_xor(sm[hh], off, 32);
        sm[hh] = 1.0f / sm[hh];
    }

    // phase 3: weighted gather; lanes own channels, serial over edges (EXEC uniform)
    float acc[8] = {0, 0, 0, 0, 0, 0, 0, 0};
    for (int j = beg; j < end; ++j) {
        int s = esrc[j];                                  // wave-uniform load
        float al[4];
#pragma unroll
        for (int hh = 0; hh < 4; ++hh) {
            float e = as_[(s << 2) + hh] + adv[hh];
            e = e > 0.0f ? e : 0.2f * e;
            al[hh] = __expf(e - mx[hh]) * sm[hh];
        }
        const float* xr = xl + (size_t)s * HC;
#pragma unroll
        for (int k = 0; k < 8; ++k)
            acc[k] = fmaf(xr[(k << 5) + lane], al[k >> 1], acc[k]);
    }

    // fused epilogue: bias + elu + residual + layernorm, in place into h
    float* hr = h + (size_t)n * HC;
    float vals[8];
    float sum = 0.0f;
#pragma unroll
    for (int k = 0; k < 8; ++k) {
        int j = (k << 5) + lane;
        float v = acc[k] + gat_b[j];
        v = v > 0.0f ? v : (__expf(v) - 1.0f);            // elu
        v += hr[j];                                       // residual (zero-padded)
        vals[k] = v; sum += v;
    }
    for (int off = 16; off > 0; off >>= 1) sum += __shfl_xor(sum, off, 32);
    float mean = sum * (1.0f / 256.0f);
    float var = 0.0f;
#pragma unroll
    for (int k = 0; k < 8; ++k) { float dd = vals[k] - mean; var += dd * dd; }
    for (int off = 16; off > 0; off >>= 1) var += __shfl_xor(var, off, 32);
    var *= (1.0f / 256.0f);
    float rs = rsqrtf(var + 1e-5f);
#pragma unroll
    for (int k = 0; k < 8; ++k) {
        int j = (k << 5) + lane;
        hr[j] = (vals[k] - mean) * rs * ln_g[j] + ln_b[j];
    }
}

// ---------------------------------------------------------------- launcher

extern "C" void kernel_launch(void* const* d_in, const int* in_sizes, int n_in,
                              void* d_out, int out_size, void* d_ws, size_t ws_size,
                              hipStream_t stream) {
    const float*     x      = (const float*)d_in[0];
    const long long* ei     = (const long long*)d_in[1];
    const float*     lin_w  = (const float*)d_in[2];
    const float*     lin_b  = (const float*)d_in[3];
    const float* gat_w[3]   = {(const float*)d_in[4],  (const float*)d_in[10], (const float*)d_in[16]};
    const float* att_src[3] = {(const float*)d_in[5],  (const float*)d_in[11], (const float*)d_in[17]};
    const float* att_dst[3] = {(const float*)d_in[6],  (const float*)d_in[12], (const float*)d_in[18]};
    const float* gat_b[3]   = {(const float*)d_in[7],  (const float*)d_in[13], (const float*)d_in[19]};
    const float* ln_g[3]    = {(const float*)d_in[8],  (const float*)d_in[14], (const float*)d_in[20]};
    const float* ln_b[3]    = {(const float*)d_in[9],  (const float*)d_in[15], (const float*)d_in[21]};
    const float* ffn_w1 = (const float*)d_in[22];
    const float* ffn_b1 = (const float*)d_in[23];
    const float* ffn_w2 = (const float*)d_in[24];
    const float* ffn_b2 = (const float*)d_in[25];
    (void)in_sizes; (void)n_in; (void)out_size; (void)ws_size;

    char* ws = (char*)d_ws;
    size_t off = 0;
    auto alloc = [&](size_t bytes) -> void* {
        void* p = ws + off;
        off += (bytes + 255) & ~(size_t)255;
        return p;
    };
    int*   src32  = (int*)  alloc((size_t)ETOT * sizeof(int));
    int*   dst32  = (int*)  alloc((size_t)ETOT * sizeof(int));
    int*   esrc   = (int*)  alloc((size_t)ETOT * sizeof(int));
    int*   deg    = (int*)  alloc((size_t)NN * sizeof(int));
    int*   rowptr = (int*)  alloc((size_t)(NN + 1) * sizeof(int));
    int*   cursor = (int*)  alloc((size_t)NN * sizeof(int));
    float* h      = (float*)alloc((size_t)NN * HC * 4);
    float* xl     = (float*)alloc((size_t)NN * HC * 4);
    float* f1     = (float*)alloc((size_t)NN * 128 * 4);
    float* as_    = (float*)alloc((size_t)NN * 4 * 4);
    float* ad_    = (float*)alloc((size_t)NN * 4 * 4);
    float* Wp     = (float*)alloc((size_t)256 * 256 * 4);

    auto cdiv = [](int a, int b) { return (a + b - 1) / b; };
    const int rowTiles = NN / 16;   // 625 exactly

    // CSR build: edges grouped by destination
    k_prep_edges<<<cdiv(ETOT, 256), 256, 0, stream>>>(ei, src32, dst32);
    k_fill_int<<<cdiv(NN, 256), 256, 0, stream>>>(deg, NN, 0);
    k_deg<<<cdiv(ETOT, 256), 256, 0, stream>>>(dst32, deg);
    k_scan<<<1, 256, 0, stream>>>(deg, rowptr, cursor);
    k_scatter<<<cdiv(ETOT, 256), 256, 0, stream>>>(src32, dst32, cursor, esrc);

    // h = relu(x @ lin_w + lin_b), stored [N,256] with zero padding (residual pad)
    k_fill<<<cdiv(NN * HC, 256), 256, 0, stream>>>(h, NN * HC, 0.0f);
    k_pack_w<<<cdiv(64 * 64, 256), 256, 0, stream>>>(lin_w, Wp, 64, 64);
    k_gemm_wmma<<<cdiv(rowTiles * 4 * 32, 256), 256, 0, stream>>>(
        x, 64, Wp, lin_b, h, HC, NN, 64, 64, 1);

    for (int l = 0; l < 3; ++l) {
        int K = (l == 0) ? 64 : HC;
        // xl = h @ gat_w[l]
        k_pack_w<<<cdiv(K * HC, 256), 256, 0, stream>>>(gat_w[l], Wp, K, HC);
        k_gemm_wmma<<<cdiv(rowTiles * 16 * 32, 256), 256, 0, stream>>>(
            h, HC, Wp, nullptr, xl, HC, NN, K, HC, 0);
        // per-(node,head) attention scalars
        k_alpha<<<cdiv(NN * 4, 256), 256, 0, stream>>>(xl, att_src[l], att_dst[l], as_, ad_);
        // fused: segment softmax + weighted aggregation + bias/elu/residual/LN
        k_gat_node<<<cdiv(NN * 32, 256), 256, 0, stream>>>(
            xl, as_, ad_, rowptr, esrc, gat_b[l], ln_g[l], ln_b[l], h);
    }

    // FFN: f1 = relu(h @ ffn_w1 + b1); out = f1 @ ffn_w2 + b2
    k_pack_w<<<cdiv(HC * 128, 256), 256, 0, stream>>>(ffn_w1, Wp, HC, 128);
    k_gemm_wmma<<<cdiv(rowTiles * 8 * 32, 256), 256, 0, stream>>>(
        h, HC, Wp, ffn_b1, f1, 128, NN, HC, 128, 1);
    k_pack_w<<<cdiv(128 * 64, 256), 256, 0, stream>>>(ffn_w2, Wp, 128, 64);
    k_gemm_wmma<<<cdiv(rowTiles * 4 * 32, 256), 256, 0, stream>>>(
        f1, 128, Wp, ffn_b2, (float*)d_out, 64, NN, 128, 64, 0);
}